// Attention_38130719654002
// MI455X (gfx1250) — compile-verified
//
#include <hip/hip_runtime.h>
#include <math.h>

typedef float v2f __attribute__((ext_vector_type(2)));
typedef float v8f __attribute__((ext_vector_type(8)));

#define BB 2
#define TT 257
#define CC 128
#define HH_ 8
#define DD_ 16
#define SP 16
#define BNS 0.9999950000374997f   // (1+1e-5)^-0.5

// ---------------- Stage A: dwconv3x3 + BN + cls concat (per q/k/v) ----------
__global__ void k_qkv_pre(const float* __restrict__ x,
                          const float* __restrict__ dw,
                          const float* __restrict__ g,
                          const float* __restrict__ bb,
                          float* __restrict__ outp)
{
    int i = blockIdx.x * blockDim.x + threadIdx.x;
    const int NT = BB * TT * CC;
    if (i >= NT) return;
    int c = i & (CC - 1);
    int t = (i >> 7) % TT;
    int b = i / (TT * CC);
    float val;
    if (t == 0) {
        val = x[i];
    } else {
        int s = t - 1;
        int yy = s >> 4, xx = s & 15;
        float acc = 0.f;
        #pragma unroll
        for (int ky = 0; ky < 3; ++ky) {
            int ny = yy + ky - 1;
            if (ny < 0 || ny >= SP) continue;
            #pragma unroll
            for (int kx = 0; kx < 3; ++kx) {
                int nx = xx + kx - 1;
                if (nx < 0 || nx >= SP) continue;
                float xv = x[((size_t)(b * TT + 1 + ny * SP + nx)) * CC + c];
                acc += xv * dw[c * 9 + ky * 3 + kx];
            }
        }
        val = acc * (g[c] * BNS) + bb[c];
    }
    outp[i] = val;
}

// ---------------- Stage B: q/k/v linear projections via WMMA f32 ------------
// out[m, n] = sum_k in[m,k] * W[n,k]   (M=514 pad->528, N=128, K=128)
// Tail guard is branch-free: clamped row + 0/1 mask multiply.
__global__ void k_qkv_gemm(const float* __restrict__ inq,
                           const float* __restrict__ ink,
                           const float* __restrict__ inv,
                           const float* __restrict__ Wq,
                           const float* __restrict__ Wk,
                           const float* __restrict__ Wv,
                           float* __restrict__ oq,
                           float* __restrict__ okk,
                           float* __restrict__ ov)
{
    const float* A; const float* W; float* O;
    int which = blockIdx.y;
    if (which == 0)      { A = inq; W = Wq; O = oq;  }
    else if (which == 1) { A = ink; W = Wk; O = okk; }
    else                 { A = inv; W = Wv; O = ov;  }
    int mt = blockIdx.x >> 3;   // 33 M tiles
    int nt = blockIdx.x & 7;    // 8  N tiles
    int lane = threadIdx.x & 31;
    int lo = lane & 15, hi = lane >> 4;
    int rowA = mt * 16 + lo;
    int rowLd = (rowA < BB * TT) ? rowA : (BB * TT - 1);
    float maskA = (rowA < BB * TT) ? 1.f : 0.f;
    int colB = nt * 16 + lo;
    const float* ap = A + (size_t)rowLd * CC;
    const float* wp = W + (size_t)colB * CC;
    v8f acc = {};
    for (int k0 = 0; k0 < CC; k0 += 4) {
        int ka = k0 + 2 * hi;
        v2f a, bf;
        a.x = ap[ka]     * maskA;
        a.y = ap[ka + 1] * maskA;
        bf.x = wp[ka];
        bf.y = wp[ka + 1];
        acc = __builtin_amdgcn_wmma_f32_16x16x4_f32(false, a, false, bf,
                                                    (short)0, acc, false, false);
    }
    #pragma unroll
    for (int r = 0; r < 8; ++r) {
        int m = mt * 16 + r + 8 * hi;
        if (m < BB * TT) O[(size_t)m * CC + nt * 16 + lo] = acc[r];
    }
}

// ---------------- Stage C: dual softmax -> score ----------------------------
__global__ void k_score(const float* __restrict__ gsc, float* __restrict__ sc)
{
    int row = blockIdx.x;   // (b*H+hh)*T + t, 4112 rows
    int tid = threadIdx.x;  // 256
    const float* gp = gsc + (size_t)row * TT;
    float z = 2.f * gp[1 + tid];
    __shared__ float sm[256];
    sm[tid] = z; __syncthreads();
    for (int s = 128; s > 0; s >>= 1) { if (tid < s) sm[tid] = fmaxf(sm[tid], sm[tid + s]); __syncthreads(); }
    float zmax = sm[0]; __syncthreads();
    sm[tid] = z; __syncthreads();
    for (int s = 128; s > 0; s >>= 1) { if (tid < s) sm[tid] = fminf(sm[tid], sm[tid + s]); __syncthreads(); }
    float zmin = sm[0]; __syncthreads();
    float e1 = expf(z - zmax);
    float e2 = expf(zmin - z);
    sm[tid] = e1; __syncthreads();
    for (int s = 128; s > 0; s >>= 1) { if (tid < s) sm[tid] += sm[tid + s]; __syncthreads(); }
    float S1 = sm[0]; __syncthreads();
    sm[tid] = e2; __syncthreads();
    for (int s = 128; s > 0; s >>= 1) { if (tid < s) sm[tid] += sm[tid + s]; __syncthreads(); }
    float S2 = sm[0];
    sc[(size_t)row * TT + 1 + tid] = 0.7f * (e1 / S1) + 0.3f * (1.f - e2 / S2);
    if (tid == 0) sc[(size_t)row * TT] = gp[0];
}

// --------- conv weight transpose, K-pair interleaved for b64 B loads --------
// Element (k, oc) of the GEMM B-matrix lives at Bm[(k>>1)*512 + oc*2 + (k&1)],
// with k = (ky*3+kx)*256 + c'. So (k, k+1) for one oc are adjacent dwords.
__global__ void k_wtr(const float* __restrict__ cw, float* __restrict__ Bm)
{
    int i = blockIdx.x * blockDim.x + threadIdx.x;  // 2304*256
    if (i >= 2304 * 256) return;
    int oc = i & 255;
    int k  = i >> 8;
    int cp = k & 255;
    int kg = k >> 8;
    int ky = kg / 3, kx = kg % 3;
    Bm[(size_t)(k >> 1) * 512 + oc * 2 + (k & 1)] =
        cw[(((size_t)oc * 256 + cp) * 3 + ky) * 3 + kx];
}

// ---------------- Stage D: implicit-GEMM stride-2 conv via WMMA f32 ---------
// One block per image n=(b,t'): M=64 positions, N=256 oc, K=2304=(ky,kx)x256.
// A built on the fly from rank-1 score x kv (9KB LDS). OOB taps read a zeroed
// 9th kv row -> branch-free zero padding.
__global__ void __launch_bounds__(512)
k_conv(const float* __restrict__ sc,    // (B,H,T,T)
       const float* __restrict__ kbuf,  // (B*T, C)
       const float* __restrict__ vbuf,
       const float* __restrict__ Bm,    // K-pair interleaved (2304,256)
       const float* __restrict__ cb,
       const float* __restrict__ g2,
       const float* __restrict__ b2,
       float* __restrict__ kvps)        // (B,H,T,65,32), entries 1..64 written
{
    __shared__ float s_s[8][256];
    __shared__ float s_kv[9][32];       // row 8 stays zero (OOB taps)
    int n = blockIdx.x;
    int b = n / TT, tp = n % TT;
    int tid = threadIdx.x;
    for (int i = tid; i < 8 * 256; i += 512) {
        int j = i >> 8, idx = i & 255;
        int m = 8 * tp + j;
        int hh = m / TT, t = m % TT;
        s_s[j][idx] = sc[(((size_t)(b * HH_ + hh)) * TT + t) * TT + 1 + idx];
    }
    for (int i = tid; i < 9 * 32; i += 512) {
        int j = i >> 5, dd = i & 31;
        if (j == 8) { s_kv[8][dd] = 0.f; continue; }
        int m = 8 * tp + j;
        int hh = m / TT, t = m % TT;
        const float* src = (dd < DD_) ? kbuf : vbuf;
        s_kv[j][dd] = src[((size_t)(b * TT + t)) * CC + hh * DD_ + (dd & 15)];
    }
    __syncthreads();

    int lane = tid & 31, lo = lane & 15, hi = lane >> 4;
    int wave = tid >> 5;
    int mtile = wave & 3;       // 4 M tiles
    int ngrp  = wave >> 2;      // 4 N groups of 4 tiles
    int pA = mtile * 16 + lo;   // output position row for A
    int oyA = pA >> 3, oxA = pA & 7;
    int ocBase = ngrp * 64 + lo;

    v8f accs[4] = {{}, {}, {}, {}};
    for (int kg = 0; kg < 9; ++kg) {
        int ky = kg / 3, kx = kg % 3;
        int y = 2 * oyA - 1 + ky;
        int x = 2 * oxA - 1 + kx;
        bool ok = (y >= 0) && (x >= 0);       // max is 15, only pad side OOB
        int js = ok ? (y >> 1) : 0;           // valid s_s row (value irrelevant)
        int jk = ok ? (y >> 1) : 8;           // zero kv row kills OOB taps
        int sb = ok ? ((y & 1) * 128 + x * 8) : 0;
        const float* srow = &s_s[js][sb];
        const float* krow = &s_kv[jk][0];
        #pragma unroll 8
        for (int c0 = 0; c0 < 256; c0 += 4) {
            int cA = c0 + 2 * hi;
            v2f a;
            a.x = srow[cA >> 5]       * krow[cA & 31];
            a.y = srow[(cA + 1) >> 5] * krow[(cA + 1) & 31];
            int k = kg * 256 + c0 + 2 * hi;   // even
            const v2f* bp = (const v2f*)(Bm + (size_t)(k >> 1) * 512);
            #pragma unroll
            for (int i = 0; i < 4; ++i) {
                v2f bf = bp[ocBase + 16 * i];
                accs[i] = __builtin_amdgcn_wmma_f32_16x16x4_f32(
                    false, a, false, bf, (short)0, accs[i], false, false);
            }
        }
    }
    // epilogue: + conv_b, BN2, scatter into kvps layout
    #pragma unroll
    for (int i = 0; i < 4; ++i) {
        int oc = ocBase + 16 * i;
        float s2 = g2[oc] * BNS;
        float cbi = cb[oc];
        float bb2 = b2[oc];
        int hh = oc >> 5, ee = oc & 31;
        size_t base = (((size_t)(b * HH_ + hh) * TT + tp) * 65) * 32 + ee;
        #pragma unroll
        for (int r = 0; r < 8; ++r) {
            int p = mtile * 16 + r + 8 * hi;
            kvps[base + (size_t)(1 + p) * 32] = (accs[i][r] + cbi) * s2 + bb2;
        }
    }
}

// ---------------- Stage E: 65-key attention + exact GELU --------------------
__global__ void k_attn(const float* __restrict__ qbuf,
                       const float* __restrict__ kbuf,
                       const float* __restrict__ vbuf,
                       const float* __restrict__ sc,
                       const float* __restrict__ kvps,
                       float* __restrict__ outp)
{
    int row = blockIdx.x;                 // (b*H+hh)*T + t
    int b  = row / (HH_ * TT);
    int rm = row % (HH_ * TT);
    int hh = rm / TT;
    int t  = rm % TT;
    int tid = threadIdx.x;                // 128
    __shared__ float lg[65];
    __shared__ float invsum;
    const float scale = 0.08838834764831845f;   // 128^-0.5
    size_t qb = ((size_t)(b * TT + t)) * CC + hh * DD_;
    float s0 = sc[(size_t)row * TT];
    if (tid < 65) {
        float acc = 0.f;
        if (tid == 0) {
            #pragma unroll
            for (int d = 0; d < DD_; ++d) acc += qbuf[qb + d] * (s0 * kbuf[qb + d]);
        } else {
            const float* kp = kvps + ((size_t)row * 65 + tid) * 32;
            #pragma unroll
            for (int d = 0; d < DD_; ++d) acc += qbuf[qb + d] * kp[d];
        }
        lg[tid] = acc * scale;
    }
    __syncthreads();
    if (tid == 0) {
        float m = -3.4e38f;
        for (int i = 0; i < 65; ++i) m = fmaxf(m, lg[i]);
        float s = 0.f;
        for (int i = 0; i < 65; ++i) { float e = expf(lg[i] - m); lg[i] = e; s += e; }
        invsum = 1.f / s;
    }
    __syncthreads();
    if (tid < DD_) {
        float o = lg[0] * (s0 * vbuf[qb + tid]);
        for (int kk = 1; kk < 65; ++kk)
            o += lg[kk] * kvps[((size_t)row * 65 + kk) * 32 + 16 + tid];
        o *= invsum;
        float ge = 0.5f * o * (1.f + erff(o * 0.70710678118654752f));
        outp[((size_t)(b * TT + t)) * CC + hh * DD_ + tid] = ge;
    }
}

extern "C" void kernel_launch(void* const* d_in, const int* in_sizes, int n_in,
                              void* d_out, int out_size, void* d_ws, size_t ws_size,
                              hipStream_t stream)
{
    (void)in_sizes; (void)n_in; (void)out_size; (void)ws_size;
    const float* x    = (const float*)d_in[0];
    const float* gsc  = (const float*)d_in[1];
    const float* dwq  = (const float*)d_in[2];
    const float* dwk  = (const float*)d_in[3];
    const float* dwv  = (const float*)d_in[4];
    const float* bnqg = (const float*)d_in[5];
    const float* bnqb = (const float*)d_in[6];
    const float* bnkg = (const float*)d_in[7];
    const float* bnkb = (const float*)d_in[8];
    const float* bnvg = (const float*)d_in[9];
    const float* bnvb = (const float*)d_in[10];
    const float* Wq   = (const float*)d_in[11];
    const float* Wk   = (const float*)d_in[12];
    const float* Wv   = (const float*)d_in[13];
    const float* cw   = (const float*)d_in[14];
    const float* cb   = (const float*)d_in[15];
    const float* g2   = (const float*)d_in[16];
    const float* b2   = (const float*)d_in[17];

    float* ws = (float*)d_ws;
    const size_t NTC = (size_t)BB * TT * CC;           // 65792
    float* qin   = ws;
    float* kin   = qin + NTC;
    float* vin   = kin + NTC;
    float* qbuf  = vin + NTC;
    float* kbuf  = qbuf + NTC;
    float* vbuf  = kbuf + NTC;
    float* score = vbuf + NTC;                          // B*H*T*T
    float* Bm    = score + (size_t)BB * HH_ * TT * TT;  // 2304*256
    float* kvps  = Bm + (size_t)2304 * 256;             // B*H*T*65*32

    int blocksA = (int)((NTC + 255) / 256);
    k_qkv_pre<<<blocksA, 256, 0, stream>>>(x, dwq, bnqg, bnqb, qin);
    k_qkv_pre<<<blocksA, 256, 0, stream>>>(x, dwk, bnkg, bnkb, kin);
    k_qkv_pre<<<blocksA, 256, 0, stream>>>(x, dwv, bnvg, bnvb, vin);

    dim3 gB(33 * 8, 3);
    k_qkv_gemm<<<gB, 32, 0, stream>>>(qin, kin, vin, Wq, Wk, Wv, qbuf, kbuf, vbuf);

    k_score<<<BB * HH_ * TT, 256, 0, stream>>>(gsc, score);

    k_wtr<<<(2304 * 256) / 256, 256, 0, stream>>>(cw, Bm);

    k_conv<<<BB * TT, 512, 0, stream>>>(score, kbuf, vbuf, Bm, cb, g2, b2, kvps);

    k_attn<<<BB * HH_ * TT, 128, 0, stream>>>(qbuf, kbuf, vbuf, score, kvps,
                                              (float*)d_out);
}